// SelectiveStateSpaceLayer_57475252355109
// MI455X (gfx1250) — compile-verified
//
#include <hip/hip_runtime.h>
#include <hip/hip_bf16.h>

typedef __attribute__((ext_vector_type(2))) float v2f;
typedef __attribute__((ext_vector_type(8))) float v8f;

#define Bsz 2
#define Lsz 2048
#define Dsz 1024
#define Nst 16
#define Rrk 16
#define Mtot (Bsz * Lsz)   // 4096 rows

// ---------------------------------------------------------------------------
// K0: Aneg[d][n] = -exp(A_log[d][n])   (1024*16 elements)
// ---------------------------------------------------------------------------
__global__ void k_aneg(const float* __restrict__ A_log, float* __restrict__ Aneg) {
    int i = blockIdx.x * 256 + threadIdx.x;
    if (i < Dsz * Nst) Aneg[i] = -__expf(A_log[i]);
}

// ---------------------------------------------------------------------------
// K1: x_dbl[m, 0:48] = x[m, :] @ Wx^T + bx   via V_WMMA_F32_16X16X4_F32
// one wave per 16-row tile; 3 column tiles (48 outputs) share the A loads
// grid = 64 blocks * 128 threads (4 waves/block), 256 row tiles total
// ---------------------------------------------------------------------------
__global__ void k_gemm1(const float* __restrict__ x, const float* __restrict__ Wx,
                        const float* __restrict__ bx, float* __restrict__ xdbl) {
    const int lane = threadIdx.x & 31;
    const int wave = threadIdx.x >> 5;
    const int tile = blockIdx.x * 4 + wave;      // 0..255
    const int row0 = tile * 16;
    const int mr   = lane & 15;                  // A row / B col within tile
    const int kg   = lane >> 4;                  // k group 0/1

    const float* xrow = x  + (size_t)(row0 + mr) * Dsz + 2 * kg;
    const float* w0   = Wx + (size_t)( 0 + mr) * Dsz + 2 * kg;
    const float* w1   = Wx + (size_t)(16 + mr) * Dsz + 2 * kg;
    const float* w2   = Wx + (size_t)(32 + mr) * Dsz + 2 * kg;

    v8f acc0 = {}; v8f acc1 = {}; v8f acc2 = {};
    for (int k0 = 0; k0 < Dsz; k0 += 4) {
        v2f a  = *(const v2f*)(xrow + k0);
        v2f b0 = *(const v2f*)(w0 + k0);
        v2f b1 = *(const v2f*)(w1 + k0);
        v2f b2 = *(const v2f*)(w2 + k0);
        acc0 = __builtin_amdgcn_wmma_f32_16x16x4_f32(false, a, false, b0, (short)0, acc0, false, false);
        acc1 = __builtin_amdgcn_wmma_f32_16x16x4_f32(false, a, false, b1, (short)0, acc1, false, false);
        acc2 = __builtin_amdgcn_wmma_f32_16x16x4_f32(false, a, false, b2, (short)0, acc2, false, false);
    }
    // D layout: element v -> row (row0 + v + 8*kg), col (lane&15)
    const int n = lane & 15;
    const int rbase = row0 + 8 * kg;
#pragma unroll
    for (int v = 0; v < 8; ++v) {
        int r = rbase + v;
        xdbl[r * 48 +  0 + n] = acc0[v] + bx[ 0 + n];
        xdbl[r * 48 + 16 + n] = acc1[v] + bx[16 + n];
        xdbl[r * 48 + 32 + n] = acc2[v] + bx[32 + n];
    }
}

// ---------------------------------------------------------------------------
// K2: dt[m, d] = softplus( dt_r[m, :] @ Wdt[d, :]^T + bdt[d] )
// K = 16 -> 4 WMMA steps. 256 M-tiles * 64 D-tiles = 16384 waves.
// grid = 2048 blocks * 256 threads (8 waves/block)
// ---------------------------------------------------------------------------
__global__ void k_gemm2(const float* __restrict__ xdbl, const float* __restrict__ Wdt,
                        const float* __restrict__ bdt, float* __restrict__ dt) {
    const int lane = threadIdx.x & 31;
    const int wave = threadIdx.x >> 5;
    const int wid  = blockIdx.x * 8 + wave;      // 0..16383
    const int mt   = wid >> 6;                   // 0..255
    const int nt   = wid & 63;                   // 0..63
    const int row0 = mt * 16;
    const int d0   = nt * 16;
    const int mr   = lane & 15;
    const int kg   = lane >> 4;

    const float* arow = xdbl + (size_t)(row0 + mr) * 48 + 2 * kg;  // dt_r = cols 0..15
    const float* brow = Wdt  + (size_t)(d0 + mr) * Rrk + 2 * kg;

    v8f acc = {};
#pragma unroll
    for (int k0 = 0; k0 < Rrk; k0 += 4) {
        v2f a = *(const v2f*)(arow + k0);
        v2f b = *(const v2f*)(brow + k0);
        acc = __builtin_amdgcn_wmma_f32_16x16x4_f32(false, a, false, b, (short)0, acc, false, false);
    }
    const int n = lane & 15;
    const int d = d0 + n;
    const float bd = bdt[d];
#pragma unroll
    for (int v = 0; v < 8; ++v) {
        int r = row0 + 8 * kg + v;
        float z  = acc[v] + bd;
        float sp = (z > 20.0f) ? z : log1pf(__expf(z));   // softplus
        dt[(size_t)r * Dsz + d] = sp;
    }
}

// ---------------------------------------------------------------------------
// K3: per position m: u = sum_d dt*x ; a[n] = sum_d exp(dt[d]*Aneg[d][n]) ;
//     b[n] = Bm[n] * u.  One 256-thread block per m; LDS tree reduce 17 slots.
// ---------------------------------------------------------------------------
__global__ void k_phase2(const float* __restrict__ x, const float* __restrict__ xdbl,
                         const float* __restrict__ dt, const float* __restrict__ Aneg,
                         float* __restrict__ aArr, float* __restrict__ bArr) {
    const int m = blockIdx.x;
    const int t = threadIdx.x;
    float u_loc = 0.0f;
    float a_loc[Nst];
#pragma unroll
    for (int n = 0; n < Nst; ++n) a_loc[n] = 0.0f;

#pragma unroll 4
    for (int d = t; d < Dsz; d += 256) {
        float dtd = dt[(size_t)m * Dsz + d];
        u_loc = fmaf(dtd, x[(size_t)m * Dsz + d], u_loc);
#pragma unroll
        for (int n = 0; n < Nst; ++n)
            a_loc[n] += __expf(dtd * Aneg[d * Nst + n]);
    }

    __shared__ float sm[17 * 256];
    sm[t] = u_loc;
#pragma unroll
    for (int n = 0; n < Nst; ++n) sm[(n + 1) * 256 + t] = a_loc[n];
    __syncthreads();
    for (int s = 128; s > 0; s >>= 1) {
        if (t < s) {
#pragma unroll
            for (int i = 0; i < 17; ++i) sm[i * 256 + t] += sm[i * 256 + t + s];
        }
        __syncthreads();
    }
    if (t < Nst) {
        float u = sm[0];
        aArr[m * Nst + t] = sm[(t + 1) * 256];
        bArr[m * Nst + t] = xdbl[m * 48 + 16 + t] * u;   // Bm * u
    }
}

// ---------------------------------------------------------------------------
// K4: sequential scan h_l = a_l*h + b_l per (b,n) chain. 32 lanes, 1 wave.
// ---------------------------------------------------------------------------
__global__ void k_scan(const float* __restrict__ aArr, const float* __restrict__ bArr,
                       float* __restrict__ hs) {
    const int lane = threadIdx.x;        // 0..31
    const int b = lane >> 4;
    const int n = lane & 15;
    float h = 0.0f;
    for (int l = 0; l < Lsz; ++l) {
        int idx = (b * Lsz + l) * Nst + n;
        h = fmaf(aArr[idx], h, bArr[idx]);
        hs[idx] = h;
    }
}

// ---------------------------------------------------------------------------
// K5: y[m] = sum_n Cm[m,n]*hs[m,n]; out[m,d] = y[m] + D_param[d]*x[m,d]
// one 256-thread block per m
// ---------------------------------------------------------------------------
__global__ void k_final(const float* __restrict__ x, const float* __restrict__ xdbl,
                        const float* __restrict__ hs, const float* __restrict__ Dp,
                        float* __restrict__ out) {
    const int m = blockIdx.x;
    const int t = threadIdx.x;
    __shared__ float sm[Nst];
    __shared__ float yv;
    if (t < Nst) sm[t] = xdbl[m * 48 + 32 + t] * hs[m * Nst + t];
    __syncthreads();
    if (t == 0) {
        float s = 0.0f;
#pragma unroll
        for (int n = 0; n < Nst; ++n) s += sm[n];
        yv = s;
    }
    __syncthreads();
    float y = yv;
#pragma unroll
    for (int i = 0; i < 4; ++i) {
        int d = t + i * 256;
        out[(size_t)m * Dsz + d] = fmaf(Dp[d], x[(size_t)m * Dsz + d], y);
    }
}

// ---------------------------------------------------------------------------
extern "C" void kernel_launch(void* const* d_in, const int* in_sizes, int n_in,
                              void* d_out, int out_size, void* d_ws, size_t ws_size,
                              hipStream_t stream) {
    const float* x     = (const float*)d_in[0];
    const float* Wx    = (const float*)d_in[1];
    const float* bx    = (const float*)d_in[2];
    const float* Wdt   = (const float*)d_in[3];
    const float* bdt   = (const float*)d_in[4];
    const float* A_log = (const float*)d_in[5];
    const float* Dp    = (const float*)d_in[6];
    float* out = (float*)d_out;

    // workspace layout (~1.6 MB); dt (16 MB) staged in d_out, overwritten by k_final
    char* ws = (char*)d_ws;
    float* xdbl = (float*)ws;                                   // 4096*48*4   = 786432
    float* Aneg = (float*)(ws + 786432);                        // 1024*16*4   = 65536
    float* aArr = (float*)(ws + 786432 + 65536);                // 4096*16*4   = 262144
    float* bArr = aArr + Mtot * Nst;                            // 262144
    float* hs   = bArr + Mtot * Nst;                            // 262144
    float* dtb  = out;                                          // 16 MB staging

    k_aneg  <<<64,   256, 0, stream>>>(A_log, Aneg);
    k_gemm1 <<<64,   128, 0, stream>>>(x, Wx, bx, xdbl);
    k_gemm2 <<<2048, 256, 0, stream>>>(xdbl, Wdt, bdt, dtb);
    k_phase2<<<Mtot, 256, 0, stream>>>(x, xdbl, dtb, Aneg, aArr, bArr);
    k_scan  <<<1,    32,  0, stream>>>(aArr, bArr, hs);
    k_final <<<Mtot, 256, 0, stream>>>(x, xdbl, hs, Dp, out);
}